// RegionL1Loss_Max_57982058496540
// MI455X (gfx1250) — compile-verified
//
#include <hip/hip_runtime.h>

#define NUM_REGIONS 116
#define NBINS 128            // padded to 128 for clean banking / WMMA chunks
#define WAVES_PER_BLOCK 8
#define BLOCK 256
#define MAX_NB 1024
#define EPSF 1e-6f
#define GAMMAF 1e-3f

typedef float v2f __attribute__((ext_vector_type(2)));
typedef float v8f __attribute__((ext_vector_type(8)));

// ---------------- Stage 1: streaming segment sums (memory-bound pass) -------
__global__ __launch_bounds__(BLOCK) void region_l1_stage1(
    const float* __restrict__ real, const float* __restrict__ fake,
    const int* __restrict__ rmap, float* __restrict__ partial, int n)
{
    // Per-wave privatized histograms: no inter-wave LDS contention.
    __shared__ float lsum[WAVES_PER_BLOCK][NBINS];
    __shared__ float lcnt[WAVES_PER_BLOCK][NBINS];

    const unsigned tid  = threadIdx.x;
    const unsigned wave = tid >> 5;          // wave32

    for (unsigned i = tid; i < WAVES_PER_BLOCK * NBINS; i += BLOCK) {
        (&lsum[0][0])[i] = 0.0f;
        (&lcnt[0][0])[i] = 0.0f;
    }
    __syncthreads();

    const unsigned stride = gridDim.x * BLOCK;
    const unsigned n4 = (unsigned)n >> 2;
    const float4* __restrict__ r4 = (const float4*)real;
    const float4* __restrict__ f4 = (const float4*)fake;
    const int4*   __restrict__ m4 = (const int4*)rmap;

    // Vectorized main loop: 3x global_load_b128 per 4 elements.
    for (unsigned i = blockIdx.x * BLOCK + tid; i < n4; i += stride) {
        float4 a = r4[i];
        float4 b = f4[i];
        int4   m = m4[i];
        float d0 = fabsf(a.x - b.x);
        float d1 = fabsf(a.y - b.y);
        float d2 = fabsf(a.z - b.z);
        float d3 = fabsf(a.w - b.w);
        atomicAdd(&lsum[wave][m.x], d0);  atomicAdd(&lcnt[wave][m.x], 1.0f);
        atomicAdd(&lsum[wave][m.y], d1);  atomicAdd(&lcnt[wave][m.y], 1.0f);
        atomicAdd(&lsum[wave][m.z], d2);  atomicAdd(&lcnt[wave][m.z], 1.0f);
        atomicAdd(&lsum[wave][m.w], d3);  atomicAdd(&lcnt[wave][m.w], 1.0f);
    }
    // Scalar tail (n not divisible by 4).
    for (unsigned i = (n4 << 2) + blockIdx.x * BLOCK + tid; i < (unsigned)n; i += stride) {
        float d = fabsf(real[i] - fake[i]);
        int r = rmap[i];
        atomicAdd(&lsum[wave][r], d);
        atomicAdd(&lcnt[wave][r], 1.0f);
    }
    __syncthreads();

    // Fold 8 wave-copies and write this block's partial row (no global atomics).
    if (tid < NUM_REGIONS) {
        float s = 0.0f, c = 0.0f;
#pragma unroll
        for (int w = 0; w < WAVES_PER_BLOCK; ++w) { s += lsum[w][tid]; c += lcnt[w][tid]; }
        float* row = partial + (size_t)blockIdx.x * (2 * NUM_REGIONS);
        row[tid]               = s;
        row[NUM_REGIONS + tid] = c;
    }
}

// ---------------- Stage 2: finalize (tiny; WMMA f32 contraction) ------------
__global__ __launch_bounds__(256) void region_l1_stage2(
    const float* __restrict__ partial, float* __restrict__ out, int n, int nblocks)
{
    __shared__ float ssum[NBINS];   // padded region sums (zeros past 116)
    __shared__ float sw[NBINS];     // padded per-region weights
    __shared__ float smean[NBINS];
    __shared__ float scnt[NBINS];
    __shared__ float smax;

    const int tid = threadIdx.x;
    if (tid < NBINS) { ssum[tid] = 0.0f; sw[tid] = 0.0f; smean[tid] = 0.0f; scnt[tid] = 0.0f; }
    __syncthreads();

    // Column-reduce block partials; lanes 0..231 are contiguous -> coalesced.
    if (tid < 2 * NUM_REGIONS) {
        float acc = 0.0f;
#pragma unroll 4
        for (int b = 0; b < nblocks; ++b)
            acc += partial[(size_t)b * (2 * NUM_REGIONS) + tid];
        if (tid < NUM_REGIONS) ssum[tid] = acc;
        else                   scnt[tid - NUM_REGIONS] = acc;
    }
    __syncthreads();

    if (tid < NUM_REGIONS)
        smean[tid] = ssum[tid] / (scnt[tid] + EPSF);
    __syncthreads();

    if (tid == 0) {
        float m = 0.0f;                      // running-max buffer starts at 0
        for (int r = 0; r < NUM_REGIONS; ++r) m = fmaxf(m, smean[r]);
        smax = m;
    }
    __syncthreads();

    if (tid < NUM_REGIONS)
        sw[tid] = 1.0f + GAMMAF * (smean[tid] / (smax + EPSF));
    __syncthreads();

    // Wave 0 (EXEC all-1s): sum_r w[r]*ssum[r] over 128 padded values using
    // two chained V_WMMA_F32_16X16X4_F32. A[i][k]=w[4i+k], B[k][j]=s[4j+k]
    // => diag(D)[i] = dot(w[4i..4i+3], s[4i..4i+3]).
    if (tid < 32) {
        const int lane = tid;
        // 16x4 f32 A layout: lanes0-15 hold K=0,1 (VGPR0,1); lanes16-31 hold K=2,3.
        const int base = ((lane & 15) << 2) + ((lane >> 4) << 1);
        v2f a0, b0, a1, b1;
        a0.x = sw[base];        a0.y = sw[base + 1];
        b0.x = ssum[base];      b0.y = ssum[base + 1];
        a1.x = sw[64 + base];   a1.y = sw[64 + base + 1];
        b1.x = ssum[64 + base]; b1.y = ssum[64 + base + 1];
        v8f c = {};
        c = __builtin_amdgcn_wmma_f32_16x16x4_f32(false, a0, false, b0, (short)0, c, false, false);
        c = __builtin_amdgcn_wmma_f32_16x16x4_f32(false, a1, false, b1, (short)0, c, false, false);

        // Diagonal extraction per C/D layout: lanes0-7 read result register number lane (M=N=lane);
        // lanes24-31 read result register number lane-24 (M=v+8, N=lane-16).
        float v = 0.0f;
#pragma unroll
        for (int j = 0; j < 8; ++j) {
            float cj = c[j];
            v += ((lane == j) || (lane == j + 24)) ? cj : 0.0f;
        }
        // wave32 reduction
        for (int off = 16; off > 0; off >>= 1)
            v += __shfl_down(v, off, 32);
        if (lane == 0)
            out[0] = v / (float)n;
    }
}

extern "C" void kernel_launch(void* const* d_in, const int* in_sizes, int n_in,
                              void* d_out, int out_size, void* d_ws, size_t ws_size,
                              hipStream_t stream) {
    const float* real = (const float*)d_in[0];
    const float* fake = (const float*)d_in[1];
    const int*   rmap = (const int*)d_in[2];
    float* out     = (float*)d_out;
    float* partial = (float*)d_ws;
    const int n = in_sizes[0];

    const size_t row_bytes = (size_t)(2 * NUM_REGIONS) * sizeof(float);
    int nb = (int)(ws_size / row_bytes);
    if (nb > MAX_NB) nb = MAX_NB;
    if (nb < 1) nb = 1;

    hipLaunchKernelGGL(region_l1_stage1, dim3(nb), dim3(BLOCK), 0, stream,
                       real, fake, rmap, partial, n);
    hipLaunchKernelGGL(region_l1_stage2, dim3(1), dim3(256), 0, stream,
                       partial, out, n, nb);
}